// QuanvolutionFilter_65481071405098
// MI455X (gfx1250) — compile-verified
//
#include <hip/hip_runtime.h>

#define THREADS          256
#define IMGS_PER_BLOCK   8
#define IMG_FLOATS       784        // 28*28
#define PATCHES_PER_IMG  196        // 14*14, also float4 chunks per image

__global__ __launch_bounds__(THREADS)
void quanv_kernel(const float* __restrict__ x, float* __restrict__ out, int nImages) {
    __shared__ float smem[IMGS_PER_BLOCK * IMG_FLOATS];   // 25088 B

    const int tid = threadIdx.x;
    const long long imgBase = (long long)blockIdx.x * IMGS_PER_BLOCK;
    int imgsHere = nImages - (int)imgBase;
    if (imgsHere > IMGS_PER_BLOCK) imgsHere = IMGS_PER_BLOCK;
    if (imgsHere <= 0) return;
    const int nChunks = imgsHere * PATCHES_PER_IMG;       // one chunk = 4 floats = 16 B

    const float* gsrc = x + imgBase * IMG_FLOATS;
    float*       gout = out + imgBase * IMG_FLOATS;       // 784 outputs per image too
    const unsigned ldsBase = (unsigned)(unsigned long long)&smem[0];

    // ---- Phase 1: contiguous async stream Global -> LDS (16 B per lane per issue) ----
    for (int c = tid; c < nChunks; c += THREADS) {
        unsigned long long ga = (unsigned long long)(gsrc + (size_t)c * 4);
        unsigned           la = ldsBase + (unsigned)c * 16u;
        asm volatile("global_load_async_to_lds_b128 %0, %1, off"
                     :: "v"(la), "v"(ga) : "memory");
    }
    asm volatile("s_wait_asynccnt 0" ::: "memory");
    __syncthreads();

    // ---- Phase 2: per-patch compute out of LDS, coalesced float4 stores ----
    const float PI_HALF = 1.57079632679489661923f;
    for (int c = tid; c < nChunks; c += THREADS) {
        int img = c / PATCHES_PER_IMG;
        int p   = c - img * PATCHES_PER_IMG;
        int i   = p / 14;
        int j   = p - i * 14;
        int row0 = img * IMG_FLOATS + (2 * i) * 28 + 2 * j;

        float2 top = *(const float2*)&smem[row0];        // ds_load_b64
        float2 bot = *(const float2*)&smem[row0 + 28];   // ds_load_b64

        float t0 = __cosf(fmaf(top.x, PI_HALF, 0.1f));
        float t1 = __cosf(fmaf(top.y, PI_HALF, 0.1f));
        float t2 = __cosf(fmaf(bot.x, PI_HALF, 0.1f));
        float t3 = __cosf(fmaf(bot.y, PI_HALF, 0.1f));

        float4 o;
        o.x = t0;
        o.y = o.x * t1;
        o.z = o.y * t2;
        o.w = o.z * t3;
        *(float4*)&gout[(size_t)c * 4] = o;              // global_store_b128
    }
}

extern "C" void kernel_launch(void* const* d_in, const int* in_sizes, int n_in,
                              void* d_out, int out_size, void* d_ws, size_t ws_size,
                              hipStream_t stream) {
    const float* x  = (const float*)d_in[0];
    float* out      = (float*)d_out;
    int nImages     = in_sizes[0] / IMG_FLOATS;           // 32768 for the reference
    int grid        = (nImages + IMGS_PER_BLOCK - 1) / IMGS_PER_BLOCK;
    quanv_kernel<<<grid, THREADS, 0, stream>>>(x, out, nImages);
}